// ContrastiveLossWithLabel_13889924235452
// MI455X (gfx1250) — compile-verified
//
#include <hip/hip_runtime.h>
#include <hip/hip_bf16.h>

// ---------------------------------------------------------------------------
// ContrastiveLossWithLabel, MI455X (gfx1250)
//
// Only 3*B entries of the BxB similarity matrix are consumed, so we never
// form it. One fused pass computes, per 16-row tile, five diagonal-of-GEMM
// chains with V_WMMA_F32_16X16X4_F32 (fp32 WMMA => reference precision):
//   ||zi||^2, ||zt||^2, zi.zt (pos), zi.zt[j1] (i2t), zi[j2].zt (t2i)
// A 16x4 per-lane layout mirrors B 4x16, so the SAME float2 load feeds both
// operands and diag(Z*W^T)[m] = dot(Z[m], W[m]).
// ---------------------------------------------------------------------------

#define B_ROWS 8192
#define DIMS   1024

typedef __attribute__((ext_vector_type(2))) float v2f;
typedef __attribute__((ext_vector_type(8))) float v8f;

__global__ __launch_bounds__(128) void contrastive_dots_kernel(
    const float* __restrict__ z_image,
    const float* __restrict__ z_text,
    const int* __restrict__ idx_i2t,
    const int* __restrict__ idx_t2i,
    float* __restrict__ ws) {
  const int tile = blockIdx.x;           // 512 tiles of 16 rows
  const int lane = threadIdx.x & 31;
  const int wave = threadIdx.x >> 5;     // 4 waves, K-slices of 256
  const int m    = lane & 15;            // matrix row handled by this lane
  const int hi   = lane >> 4;            // 0 -> K=0,1 ; 1 -> K=2,3
  const int row  = tile * 16 + m;

  const int j1 = idx_i2t[row];           // gathered z_text row for i2t
  const int j2 = idx_t2i[row];           // gathered z_image row for t2i

  const float* pi = z_image + (size_t)row * DIMS;
  const float* pt = z_text  + (size_t)row * DIMS;
  const float* pg = z_text  + (size_t)j1  * DIMS;
  const float* ph = z_image + (size_t)j2  * DIMS;

  // gfx1250 prefetch path (global_prefetch_b8) to warm caches for this slice
  __builtin_prefetch(pi + wave * 256, 0, 3);
  __builtin_prefetch(pt + wave * 256, 0, 3);
  __builtin_prefetch(pg + wave * 256, 0, 3);
  __builtin_prefetch(ph + wave * 256, 0, 3);

  v8f c_ii = {};  // diag -> ||zi||^2
  v8f c_tt = {};  // diag -> ||zt||^2
  v8f c_it = {};  // diag -> zi . zt          (pos, raw)
  v8f c_ig = {};  // diag -> zi . zt[j1]      (i2t, raw)
  v8f c_ht = {};  // diag -> zi[j2] . zt      (t2i, raw)

  int k0 = wave * 256;
#pragma unroll 4
  for (int t = 0; t < 64; ++t, k0 += 4) {
    const int k = k0 + 2 * hi;           // lane's K pair within this step
    v2f a_i = *(const v2f*)(pi + k);
    v2f a_t = *(const v2f*)(pt + k);
    v2f b_g = *(const v2f*)(pg + k);
    v2f a_h = *(const v2f*)(ph + k);
    // (neg_a, A, neg_b, B, c_mod, C, reuse_a, reuse_b)
    c_ii = __builtin_amdgcn_wmma_f32_16x16x4_f32(false, a_i, false, a_i, (short)0, c_ii, false, false);
    c_tt = __builtin_amdgcn_wmma_f32_16x16x4_f32(false, a_t, false, a_t, (short)0, c_tt, false, false);
    c_it = __builtin_amdgcn_wmma_f32_16x16x4_f32(false, a_i, false, a_t, (short)0, c_it, false, false);
    c_ig = __builtin_amdgcn_wmma_f32_16x16x4_f32(false, a_i, false, b_g, (short)0, c_ig, false, false);
    c_ht = __builtin_amdgcn_wmma_f32_16x16x4_f32(false, a_h, false, a_t, (short)0, c_ht, false, false);
  }

  // Diagonal extraction per C/D layout: element (m,m) lives at
  //   m in 0..7  -> lane m,      vgpr m
  //   m in 8..15 -> lane m + 16, vgpr m - 8
  __shared__ float red[4][5][16];
  const bool valid = (lane < 8) || (lane >= 24);
  const int  dm    = (lane < 8) ? lane : (lane - 16);  // which diag row
  const int  vi    = lane & 7;                         // which accumulator vgpr
  if (valid) {
    red[wave][0][dm] = c_ii[vi];
    red[wave][1][dm] = c_tt[vi];
    red[wave][2][dm] = c_it[vi];
    red[wave][3][dm] = c_ig[vi];
    red[wave][4][dm] = c_ht[vi];
  }
  __syncthreads();

  const int t = threadIdx.x;
  if (t < 80) {                          // 5 quantities x 16 rows
    const int c  = t >> 4;
    const int mm = t & 15;
    const float s = red[0][c][mm] + red[1][c][mm] + red[2][c][mm] + red[3][c][mm];
    ws[(size_t)c * B_ROWS + tile * 16 + mm] = s;
  }
}

__global__ __launch_bounds__(256) void contrastive_finalize_kernel(
    const float* __restrict__ ws,
    const int* __restrict__ idx_i2t,
    const int* __restrict__ idx_t2i,
    float* __restrict__ out) {
  const int i = blockIdx.x * blockDim.x + threadIdx.x;
  float contrib = 0.0f;
  if (i < B_ROWS) {
    const float* ni2  = ws;
    const float* nt2  = ws + 1 * B_ROWS;
    const float* posr = ws + 2 * B_ROWS;
    const float* i2tr = ws + 3 * B_ROWS;
    const float* t2ir = ws + 4 * B_ROWS;

    const float eps  = 1e-12f;
    const float invT = 1.0f / 0.07f;

    const int j1 = idx_i2t[i];
    const int j2 = idx_t2i[i];

    const float rni   = 1.0f / fmaxf(sqrtf(ni2[i]),  eps);
    const float rnt   = 1.0f / fmaxf(sqrtf(nt2[i]),  eps);
    const float rnt_g = 1.0f / fmaxf(sqrtf(nt2[j1]), eps);
    const float rni_g = 1.0f / fmaxf(sqrtf(ni2[j2]), eps);

    const float pos = posr[i] * rni   * rnt   * invT;
    const float n1  = i2tr[i] * rni   * rnt_g * invT;
    const float n2  = t2ir[i] * rni_g * rnt   * invT;

    const float x1 = n1 - pos;
    const float x2 = n2 - pos;
    // stable softplus: max(x,0) + log1p(exp(-|x|))
    const float sp1 = fmaxf(x1, 0.0f) + log1pf(__expf(-fabsf(x1)));
    const float sp2 = fmaxf(x2, 0.0f) + log1pf(__expf(-fabsf(x2)));
    contrib = 0.5f * (sp1 + sp2) * (1.0f / (float)B_ROWS);
  }

  __shared__ float sdata[256];
  sdata[threadIdx.x] = contrib;
  __syncthreads();
  for (int s = 128; s > 0; s >>= 1) {
    if (threadIdx.x < s) sdata[threadIdx.x] += sdata[threadIdx.x + s];
    __syncthreads();
  }
  if (threadIdx.x == 0) atomicAdd(out, sdata[0]);
}

extern "C" void kernel_launch(void* const* d_in, const int* in_sizes, int n_in,
                              void* d_out, int out_size, void* d_ws, size_t ws_size,
                              hipStream_t stream) {
  const float* z_image = (const float*)d_in[0];
  const float* z_text  = (const float*)d_in[1];
  // d_in[2] = labels (unused by the loss itself)
  const int* idx_i2t   = (const int*)d_in[3];
  const int* idx_t2i   = (const int*)d_in[4];
  float* out           = (float*)d_out;
  float* ws            = (float*)d_ws;   // 5*B floats = 160 KB

  hipMemsetAsync(out, 0, sizeof(float), stream);

  contrastive_dots_kernel<<<B_ROWS / 16, 128, 0, stream>>>(
      z_image, z_text, idx_i2t, idx_t2i, ws);

  contrastive_finalize_kernel<<<B_ROWS / 256, 256, 0, stream>>>(
      ws, idx_i2t, idx_t2i, out);
}